// Osci_AE_13202729468542
// MI455X (gfx1250) — compile-verified
//
#include <hip/hip_runtime.h>

// Kuramoto oscillator net, sparse LDS-persistent formulation.
// One workgroup per batch element; phases live in LDS across all T steps;
// gathered sparse dot products evaluated on the matrix pipe via
// V_WMMA_F32_16X16X4_F32 (diagonal extraction).
// sin/cos interleaved as float2 in LDS so every neighbor gather is one
// ds_load_b64; per-row 1/n precomputed so the hot loop has no division.

typedef __attribute__((ext_vector_type(2))) float v2f;
typedef __attribute__((ext_vector_type(8))) float v8f;

#define N_OSC  2048
#define KN     8
#define BATCH  32
#define TSTEPS 20

__device__ __forceinline__ float diag_sel(v8f a, int d) {
  // pick component d (0..7); lane-dependent but branch-free
  float r = a[0];
  r = (d == 1) ? a[1] : r;
  r = (d == 2) ? a[2] : r;
  r = (d == 3) ? a[3] : r;
  r = (d == 4) ? a[4] : r;
  r = (d == 5) ? a[5] : r;
  r = (d == 6) ? a[6] : r;
  r = (d == 7) ? a[7] : r;
  return r;
}

__global__ __launch_bounds__(1024, 1)
void kuramoto_wmma(const float* __restrict__ coupling,   // [B,N,K]
                   const float* __restrict__ phase0,     // [B,N]
                   const float* __restrict__ omega,      // [B,N]
                   const int*   __restrict__ conn,       // [N,K]
                   float*       __restrict__ out) {      // [T+1,B,N]
  __shared__ float2 sc_sh[N_OSC];     // interleaved {sin, cos} -> b64 gathers
  __shared__ float  ph_sh[N_OSC];
  __shared__ float  om_sh[N_OSC];
  __shared__ float  rn_sh[N_OSC];     // 1 / nnz(row)

  const int b    = blockIdx.x;
  const int tid  = threadIdx.x;
  const int lane = tid & 31;          // wave32
  const int wave = tid >> 5;          // 32 waves per block
  const int half = lane >> 4;         // half-wave selects K pair in A/B layout
  const int j    = lane & 15;         // tile row (A: M) == tile col (B: N)
  const int waveBase = wave * 64;     // 64 oscillators (4 tiles of 16) per wave

  // ---- init: phases/omega -> LDS, out[0] = phase0, per-row 1/nnz ----
  for (int i = tid; i < N_OSC; i += 1024) {
    float p  = phase0[b * N_OSC + i];
    ph_sh[i] = p;
    om_sh[i] = omega[b * N_OSC + i];
    out[b * N_OSC + i] = p;                       // t = 0 slab

    int   idxr[KN];
    float wr[KN];
#pragma unroll
    for (int k = 0; k < KN; ++k) {
      idxr[k] = conn[i * KN + k];
      wr[k]   = coupling[(b * N_OSC + i) * KN + k];
    }
    // dense scatter semantics: duplicate column -> last write wins;
    // n = count of nonzero surviving entries (constant all run -> store 1/n)
    int n = 0;
#pragma unroll
    for (int k = 0; k < KN; ++k) {
      bool keep = true;
#pragma unroll
      for (int kk = k + 1; kk < KN; ++kk) keep = keep && (idxr[kk] != idxr[k]);
      if (keep && wr[k] != 0.0f) n++;
    }
    rn_sh[i] = 1.0f / (float)n;       // one divide at setup, none in hot loop
  }

  // ---- per-lane WMMA-layout A operands (dedup-masked coupling) + gather idx
  // A 16x4 f32 layout: VGPR0 = {K=0 | K=2}, VGPR1 = {K=1 | K=3} across halves;
  // K-hi tile uses K+4. B (4x16) mirrors this with lanes = N.
  float aw[4][4];
  int   ai[4][4];
#pragma unroll
  for (int g = 0; g < 4; ++g) {
    const int row = waveBase + g * 16 + j;
    int   idxr[KN];
    float wr[KN];
#pragma unroll
    for (int k = 0; k < KN; ++k) {
      idxr[k] = conn[row * KN + k];
      wr[k]   = coupling[(b * N_OSC + row) * KN + k];
    }
#pragma unroll
    for (int k = 0; k < KN; ++k) {
      bool keep = true;
#pragma unroll
      for (int kk = k + 1; kk < KN; ++kk) keep = keep && (idxr[kk] != idxr[k]);
      if (!keep) wr[k] = 0.0f;                   // last duplicate wins
    }
    const int k0 = 2 * half;
    aw[g][0] = wr[k0];     ai[g][0] = idxr[k0];
    aw[g][1] = wr[k0 + 1]; ai[g][1] = idxr[k0 + 1];
    aw[g][2] = wr[k0 + 4]; ai[g][2] = idxr[k0 + 4];
    aw[g][3] = wr[k0 + 5]; ai[g][3] = idxr[k0 + 5];
  }

  __syncthreads();

  // diagonal-owning lanes: (M=d,N=d) lives at lane d (d<8) / lane d+16 (d>=8),
  // accumulator component d&7
  const bool upd  = (lane < 8) || (lane >= 24);
  const int  dloc = (lane < 8) ? lane : (lane - 16);
  const int  dsel = lane & 7;

  for (int t = 0; t < TSTEPS; ++t) {
    // (a) sin/cos of current phases (VALU/trans pipe), one b64 store each
    for (int i = tid; i < N_OSC; i += 1024) {
      float sv, cv;
      __sincosf(ph_sh[i], &sv, &cv);
      sc_sh[i] = make_float2(sv, cv);
    }
    __syncthreads();

    // (b) gathered sparse dots on the matrix pipe; EPS=1.0, ANNEAL=0
#pragma unroll
    for (int g = 0; g < 4; ++g) {
      const float2 p0 = sc_sh[ai[g][0]];   // one ds_load_b64 per neighbor
      const float2 p1 = sc_sh[ai[g][1]];
      const float2 p2 = sc_sh[ai[g][2]];
      const float2 p3 = sc_sh[ai[g][3]];

      v2f a_lo, a_hi, bs_lo, bs_hi, bc_lo, bc_hi;
      a_lo[0] = aw[g][0]; a_lo[1] = aw[g][1];
      a_hi[0] = aw[g][2]; a_hi[1] = aw[g][3];
      bs_lo[0] = p0.x; bs_lo[1] = p1.x;
      bs_hi[0] = p2.x; bs_hi[1] = p3.x;
      bc_lo[0] = p0.y; bc_lo[1] = p1.y;
      bc_hi[0] = p2.y; bc_hi[1] = p3.y;

      v8f z = {};
      v8f accs = __builtin_amdgcn_wmma_f32_16x16x4_f32(
          false, a_lo, false, bs_lo, (short)0, z, false, false);
      accs = __builtin_amdgcn_wmma_f32_16x16x4_f32(
          false, a_hi, false, bs_hi, (short)0, accs, false, false);
      v8f accc = __builtin_amdgcn_wmma_f32_16x16x4_f32(
          false, a_lo, false, bc_lo, (short)0, z, false, false);
      accc = __builtin_amdgcn_wmma_f32_16x16x4_f32(
          false, a_hi, false, bc_hi, (short)0, accc, false, false);

      if (upd) {
        const int osc = waveBase + g * 16 + dloc;
        const float Cs = diag_sel(accs, dsel);
        const float Cc = diag_sel(accc, dsel);
        const float2 scp = sc_sh[osc];     // own {sin,cos}, one b64 load
        const float delta = (Cs * scp.y - Cc * scp.x) * rn_sh[osc];
        const float pn = ph_sh[osc] + delta + om_sh[osc];
        ph_sh[osc] = pn;
        out[(t + 1) * (BATCH * N_OSC) + b * N_OSC + osc] = pn;
      }
    }
    __syncthreads();
  }
}

extern "C" void kernel_launch(void* const* d_in, const int* in_sizes, int n_in,
                              void* d_out, int out_size, void* d_ws, size_t ws_size,
                              hipStream_t stream) {
  (void)in_sizes; (void)n_in; (void)out_size; (void)d_ws; (void)ws_size;
  const float* coupling = (const float*)d_in[0];   // [B,N,K] f32
  const float* phase0   = (const float*)d_in[1];   // [B,N]   f32
  const float* omega    = (const float*)d_in[2];   // [B,N]   f32
  const int*   conn     = (const int*)d_in[3];     // [N,K]   i32
  float* out = (float*)d_out;                      // [T+1,B,N] f32

  kuramoto_wmma<<<BATCH, 1024, 0, stream>>>(coupling, phase0, omega, conn, out);
}